// LSTM_39187281609127
// MI455X (gfx1250) — compile-verified
//
#include <hip/hip_runtime.h>
#include <hip/hip_bf16.h>

// ---------------------------------------------------------------------------
// Fused 2-layer LSTM (B=2048, T=4096, IN=1, H=8, OUT=1) for gfx1250 (MI455X).
//
// One wave32 owns 32 batch rows for the whole T=4096 recurrence.
// Recurrent GEMMs computed as gates^T = W * h^T with V_WMMA_F32_16X16X4_F32:
//   A (weights)   : constant across all timesteps, resident in VGPRs
//   B (hidden)    : rebuilt per step from state regs via v_permlanex16 + select
//   C (input+bias): fused into the WMMA accumulator chain
// Cross-lane half-wave swaps use v_permlanex16_b32 (pure VALU) instead of
// ds_bpermute, keeping the DS pipe and DScnt waits off the serial critical
// path. No LDS, no HBM traffic besides the x stream and the output stream.
// ---------------------------------------------------------------------------

typedef float v2f __attribute__((ext_vector_type(2)));
typedef float v8f __attribute__((ext_vector_type(8)));

#define Bsz 2048
#define Tsz 4096
#define Hsz 8

// Swap the two 16-lane halves of the wave (xor-16 shuffle) as a VALU op.
__device__ __forceinline__ float shx16(float v) {
#if __has_builtin(__builtin_amdgcn_permlanex16)
    // identity lane selects -> dest lane i reads source lane i of the other
    // 16-lane half: exactly __shfl_xor(v, 16) for wave32, but on the VALU
    // (v_permlanex16_b32) instead of the LDS pipe (ds_bpermute_b32).
    unsigned int u = __float_as_uint(v);
    unsigned int r = __builtin_amdgcn_permlanex16(
        u, u, 0x76543210u, 0xfedcba98u, /*fi=*/false, /*bound_ctrl=*/false);
    return __uint_as_float(r);
#else
    return __shfl_xor(v, 16, 32);
#endif
}

__device__ __forceinline__ float fsig(float x) {
    // 1/(1+e^-x): v_exp_f32 + v_rcp_f32 (TRANS, co-issues with WMMA)
    return __builtin_amdgcn_rcpf(1.0f + __expf(-x));
}
__device__ __forceinline__ float ftanh(float x) {
    // tanh(x) = 1 - 2/(e^{2x}+1)
    return 1.0f - 2.0f * __builtin_amdgcn_rcpf(__expf(2.0f * x) + 1.0f);
}

__device__ __forceinline__ v8f wmma4(v2f a, v2f b, v8f c) {
    // D = A(16x4 f32) * B(4x16 f32) + C(16x16 f32)
    return __builtin_amdgcn_wmma_f32_16x16x4_f32(
        /*neg_a=*/false, a, /*neg_b=*/false, b,
        /*c_mod=*/(short)0, c, /*reuse_a=*/false, /*reuse_b=*/false);
}

__global__ __launch_bounds__(32) void lstm2_wmma_kernel(
    const float* __restrict__ x,     // [B,T,1]
    const float* __restrict__ Wih0,  // [32,1]
    const float* __restrict__ Whh0,  // [32,8]
    const float* __restrict__ bih0, const float* __restrict__ bhh0,  // [32]
    const float* __restrict__ Wih1,  // [32,8]
    const float* __restrict__ Whh1,  // [32,8]
    const float* __restrict__ bih1, const float* __restrict__ bhh1,  // [32]
    const float* __restrict__ Wlin,  // [1,8]
    const float* __restrict__ blin,  // [1]
    float* __restrict__ out)         // [B,T,1]
{
    const int lane = threadIdx.x;        // 0..31  (wave32)
    const int lm   = lane & 15;          // row within 16-wide tile
    const bool hh  = (lane >= 16);       // upper lane half
    const int hsel = hh ? 1 : 0;
    const long row = (long)(blockIdx.x * 32 + lane) * Tsz;  // x/out row base

    // ---- Preload constant A fragments (weights, gates^T orientation) ------
    // A tile layout (16x4 f32): lane m: V0=A(m,k0+0),V1=A(m,k0+1);
    //                           lane m+16: V0=A(m,k0+2),V1=A(m,k0+3)
    v2f A0[2][2];      // layer0: [gate_tile][k_frag], K=8
    v2f A1[2][4];      // layer1: [gate_tile][k_frag], K=16 = [Wih1 | Whh1]
#pragma unroll
    for (int gt = 0; gt < 2; ++gt) {
        const int g = 16 * gt + lm;
#pragma unroll
        for (int kf = 0; kf < 2; ++kf) {
            const int kb = 4 * kf + 2 * hsel;
            A0[gt][kf].x = Whh0[g * Hsz + kb];
            A0[gt][kf].y = Whh0[g * Hsz + kb + 1];
        }
#pragma unroll
        for (int kf = 0; kf < 4; ++kf) {
            const int k = 4 * kf + 2 * hsel;
            const float* Wsrc = (kf < 2) ? Wih1 : Whh1;
            const int kk = (kf < 2) ? k : (k - 8);
            A1[gt][kf].x = Wsrc[g * Hsz + kk];
            A1[gt][kf].y = Wsrc[g * Hsz + kk + 1];
        }
    }

    // ---- Per-lane constants for the C (input+bias) tiles ------------------
    // C/D layout: VGPR r: lanes0-15 -> gate row 16gt+r, lanes16-31 -> 16gt+r+8
    float wg0C[2][8], b0C[2][8], b1C[2][8];
#pragma unroll
    for (int gt = 0; gt < 2; ++gt)
#pragma unroll
        for (int r = 0; r < 8; ++r) {
            const int g = 16 * gt + r + 8 * hsel;
            wg0C[gt][r] = Wih0[g];               // IN == 1
            b0C[gt][r]  = bih0[g] + bhh0[g];
            b1C[gt][r]  = bih1[g] + bhh1[g];
        }

    float wl[8];
#pragma unroll
    for (int j = 0; j < 8; ++j) wl[j] = Wlin[j];
    const float bl = blin[0];

    // ---- Recurrent state: h(j, batch=lane) in lane-parallel registers -----
    float h1[8], c1[8], h2[8], c2[8];
#pragma unroll
    for (int j = 0; j < 8; ++j) { h1[j] = 0.f; c1[j] = 0.f; h2[j] = 0.f; c2[j] = 0.f; }

    for (int t = 0; t < Tsz; ++t) {
        // x(batch=lane) at time t; stream is per-lane sequential in t.
        const float xv = x[row + t];
        __builtin_prefetch(&x[row + t + 256], 0, 0);   // global_prefetch_b8
        const float xsh = shx16(xv);
        const float xb0 = hh ? xsh : xv;   // batches 0..15 in both halves
        const float xb1 = hh ? xv : xsh;   // batches 16..31 in both halves

        // ---- B fragments (4x16 f32) of h1^T for layer-0 recurrent GEMM ----
        // B layout: V0: lanes0-15 K=k0,N=lane; lanes16-31 K=k0+2; V1: k0+1/k0+3
        v2f Bh1[2][2];  // [k_frag][batch_tile]
#pragma unroll
        for (int kf = 0; kf < 2; ++kf) {
            const int k0 = 4 * kf;
            const float s0 = shx16(h1[k0 + 0]), s1 = shx16(h1[k0 + 1]);
            const float s2 = shx16(h1[k0 + 2]), s3 = shx16(h1[k0 + 3]);
            Bh1[kf][0].x = hh ? s2 : h1[k0 + 0];
            Bh1[kf][0].y = hh ? s3 : h1[k0 + 1];
            Bh1[kf][1].x = hh ? h1[k0 + 2] : s0;
            Bh1[kf][1].y = hh ? h1[k0 + 3] : s1;
        }

        // ---- Layer 0: gates^T = Whh0 * h1^T + (x*Wih0 + b) ---------------
        v8f D0[2][2];
#pragma unroll
        for (int gt = 0; gt < 2; ++gt)
#pragma unroll
            for (int bt = 0; bt < 2; ++bt) {
                const float xb = bt ? xb1 : xb0;
                v8f acc;
#pragma unroll
                for (int r = 0; r < 8; ++r)
                    acc[r] = fmaf(xb, wg0C[gt][r], b0C[gt][r]);
                acc = wmma4(A0[gt][0], Bh1[0][bt], acc);   // K = 0..3
                acc = wmma4(A0[gt][1], Bh1[1][bt], acc);   // K = 4..7
                D0[gt][bt] = acc;
            }

        // ---- Layer-0 gate nonlinearities + state update -------------------
#pragma unroll
        for (int r = 0; r < 8; ++r) {
            const float a00 = D0[0][0][r], a01 = D0[0][1][r];
            const float a10 = D0[1][0][r], a11 = D0[1][1][r];
            const float sA0 = shx16(a00), sB0 = shx16(a01);
            const float sA1 = shx16(a10), sB1 = shx16(a11);
            const float ig = fsig(hh ? sB0 : a00);   // rows  0..7  : i
            const float fg = fsig(hh ? a01 : sA0);   // rows  8..15 : f
            const float gg = ftanh(hh ? sB1 : a10);  // rows 16..23 : g
            const float og = fsig(hh ? a11 : sA1);   // rows 24..31 : o
            c1[r] = fmaf(fg, c1[r], ig * gg);
            h1[r] = og * ftanh(c1[r]);
        }

        // ---- B fragments for layer 1: K = [h1_t (0..7) | h2_prev (8..15)] -
        v2f Bc[4][2];
#pragma unroll
        for (int kf = 0; kf < 4; ++kf) {
            const float* hs = (kf < 2) ? h1 : h2;
            const int k0 = 4 * (kf & 1);
            const float s0 = shx16(hs[k0 + 0]), s1 = shx16(hs[k0 + 1]);
            const float s2 = shx16(hs[k0 + 2]), s3 = shx16(hs[k0 + 3]);
            Bc[kf][0].x = hh ? s2 : hs[k0 + 0];
            Bc[kf][0].y = hh ? s3 : hs[k0 + 1];
            Bc[kf][1].x = hh ? hs[k0 + 2] : s0;
            Bc[kf][1].y = hh ? hs[k0 + 3] : s1;
        }

        // ---- Layer 1: gates^T = [Wih1|Whh1] * [h1;h2]^T + b ---------------
        v8f D1[2][2];
#pragma unroll
        for (int gt = 0; gt < 2; ++gt)
#pragma unroll
            for (int bt = 0; bt < 2; ++bt) {
                v8f acc;
#pragma unroll
                for (int r = 0; r < 8; ++r) acc[r] = b1C[gt][r];
                acc = wmma4(A1[gt][0], Bc[0][bt], acc);    // K  0..3  (Wih1)
                acc = wmma4(A1[gt][1], Bc[1][bt], acc);    // K  4..7  (Wih1)
                acc = wmma4(A1[gt][2], Bc[2][bt], acc);    // K  8..11 (Whh1)
                acc = wmma4(A1[gt][3], Bc[3][bt], acc);    // K 12..15 (Whh1)
                D1[gt][bt] = acc;
            }

        // ---- Layer-1 gate nonlinearities + state update -------------------
#pragma unroll
        for (int r = 0; r < 8; ++r) {
            const float a00 = D1[0][0][r], a01 = D1[0][1][r];
            const float a10 = D1[1][0][r], a11 = D1[1][1][r];
            const float sA0 = shx16(a00), sB0 = shx16(a01);
            const float sA1 = shx16(a10), sB1 = shx16(a11);
            const float ig = fsig(hh ? sB0 : a00);
            const float fg = fsig(hh ? a01 : sA0);
            const float gg = ftanh(hh ? sB1 : a10);
            const float og = fsig(hh ? a11 : sA1);
            c2[r] = fmaf(fg, c2[r], ig * gg);
            h2[r] = og * ftanh(c2[r]);
        }

        // ---- Output projection: pred(batch=lane) = h2 . Wlin + b ----------
        float p = bl;
#pragma unroll
        for (int j = 0; j < 8; ++j) p = fmaf(h2[j], wl[j], p);
        out[row + t] = p;
    }
}

extern "C" void kernel_launch(void* const* d_in, const int* in_sizes, int n_in,
                              void* d_out, int out_size, void* d_ws, size_t ws_size,
                              hipStream_t stream) {
    (void)in_sizes; (void)n_in; (void)out_size; (void)d_ws; (void)ws_size;
    const float* x    = (const float*)d_in[0];
    const float* Wih0 = (const float*)d_in[1];
    const float* Whh0 = (const float*)d_in[2];
    const float* bih0 = (const float*)d_in[3];
    const float* bhh0 = (const float*)d_in[4];
    const float* Wih1 = (const float*)d_in[5];
    const float* Whh1 = (const float*)d_in[6];
    const float* bih1 = (const float*)d_in[7];
    const float* bhh1 = (const float*)d_in[8];
    const float* Wlin = (const float*)d_in[9];
    const float* blin = (const float*)d_in[10];
    float* out = (float*)d_out;

    // 2048 batches / 32 per wave = 64 workgroups of one wave32 each.
    lstm2_wmma_kernel<<<Bsz / 32, 32, 0, stream>>>(
        x, Wih0, Whh0, bih0, bhh0, Wih1, Whh1, bih1, bhh1, Wlin, blin, out);
}